// DSGraphGLayer_58926951301764
// MI455X (gfx1250) — compile-verified
//
#include <hip/hip_runtime.h>

typedef __attribute__((ext_vector_type(2))) float v2f;
typedef __attribute__((ext_vector_type(8))) float v8f;

#define N_SET   512
#define M_NODES 256
#define DD      128
#define NM      (N_SET * M_NODES)          // 131072 nodes
#define NEDGE   1048576

// ---------------------------------------------------------------------------
// Degree pipeline: deg starts at 1 (self loop), add 1 per incoming edge,
// then dinv = rsqrt(deg) in place.
// ---------------------------------------------------------------------------
__global__ void k_deg_init(float* __restrict__ deg) {
    int i = blockIdx.x * blockDim.x + threadIdx.x;
    if (i < NM) deg[i] = 1.0f;
}

__global__ void k_deg_edges(const int* __restrict__ dst, float* __restrict__ deg) {
    int e = blockIdx.x * blockDim.x + threadIdx.x;
    if (e < NEDGE) atomicAdd(&deg[dst[e]], 1.0f);
}

__global__ void k_dinv(float* __restrict__ deg) {
    int i = blockIdx.x * blockDim.x + threadIdx.x;
    if (i < NM) deg[i] = rsqrtf(deg[i]);
}

// ---------------------------------------------------------------------------
// Fused dense stage: H[(n,m),:] = x[n,m,:]@W0 + sum_i (A_i x)[n,m,:]@W[i+1]
// One wave (32 threads) per (nb, m): 16 n-rows x 128 d-cols output tile.
// Uses V_WMMA_F32_16X16X4_F32 throughout (full fp32 precision).
//
// Fragment layouts (CDNA5 ISA 7.12.2, wave32):
//   A 16x4 : lane(lrow,half): a0=A[lrow][2h], a1=A[lrow][2h+1]
//   B 4x16 : lane(lcol,half): b0=B[2h][lcol], b1=B[2h+1][lcol]
//   C/D    : vgpr v of lane(lcol,half) = D[v+8h][lcol]
// ---------------------------------------------------------------------------
__global__ __launch_bounds__(32) void k_gemm_fused(
        const float* __restrict__ x,      // [512,256,128]
        const float* __restrict__ cadj,   // [2,512,512]
        const float* __restrict__ W,      // [3,128,128]
        float* __restrict__ H)            // [131072,128]
{
    __shared__ float Zt[16][132];         // padded: bank-conflict-free
    const int lane = threadIdx.x;
    const int lrow = lane & 15;
    const int half = lane >> 4;
    const int m  = blockIdx.x;            // 0..255
    const int nb = blockIdx.y;            // 0..31

    // ---- stage Z0 = x rows (n = nb*16..+15, fixed m) into LDS
    #pragma unroll
    for (int it = 0; it < 16; ++it) {
        int idx = it * 32 + lane;         // 512 float4 total
        int r   = idx >> 5;
        int c4  = idx & 31;
        float4 vv = *(const float4*)(x + ((size_t)(nb * 16 + r) * M_NODES + m) * DD + c4 * 4);
        *(float4*)(&Zt[r][c4 * 4]) = vv;
    }
    __syncthreads();

    v8f Hacc[8];
    #pragma unroll
    for (int t = 0; t < 8; ++t) Hacc[t] = (v8f){};

    // ---- Hacc += Z0 @ W0
    {
        const float* Wm = W;
        for (int k0 = 0; k0 < DD; k0 += 4) {
            v2f a = *(const v2f*)(&Zt[lrow][k0 + 2 * half]);
            const float* w0 = Wm + (size_t)(k0 + 2 * half) * DD;
            #pragma unroll
            for (int t = 0; t < 8; ++t) {
                v2f bf; bf[0] = w0[t * 16 + lrow]; bf[1] = w0[DD + t * 16 + lrow];
                Hacc[t] = __builtin_amdgcn_wmma_f32_16x16x4_f32(
                    false, a, false, bf, (short)0, Hacc[t], false, false);
            }
        }
    }

    // ---- per hop: Y = A_hop @ X  (K = 512 over set dim), then Hacc += Y @ W[hop+1]
    for (int hop = 0; hop < 2; ++hop) {
        v8f Yacc[8];
        #pragma unroll
        for (int t = 0; t < 8; ++t) Yacc[t] = (v8f){};

        const float* Ar = cadj + (size_t)hop * N_SET * N_SET
                               + (size_t)(nb * 16 + lrow) * N_SET;
        for (int j0 = 0; j0 < N_SET; j0 += 4) {
            v2f a = *(const v2f*)(Ar + j0 + 2 * half);
            const float* xr = x + (size_t)(j0 + 2 * half) * (M_NODES * DD) + (size_t)m * DD;
            #pragma unroll
            for (int t = 0; t < 8; ++t) {
                v2f bf;
                bf[0] = xr[t * 16 + lrow];
                bf[1] = xr[(size_t)M_NODES * DD + t * 16 + lrow];
                Yacc[t] = __builtin_amdgcn_wmma_f32_16x16x4_f32(
                    false, a, false, bf, (short)0, Yacc[t], false, false);
            }
        }

        // D-layout -> LDS so it can be re-read in A-layout
        __syncthreads();
        #pragma unroll
        for (int t = 0; t < 8; ++t)
            #pragma unroll
            for (int v = 0; v < 8; ++v)
                Zt[v + 8 * half][t * 16 + lrow] = Yacc[t][v];
        __syncthreads();

        const float* Wm = W + (size_t)(hop + 1) * DD * DD;
        for (int k0 = 0; k0 < DD; k0 += 4) {
            v2f a = *(const v2f*)(&Zt[lrow][k0 + 2 * half]);
            const float* w0 = Wm + (size_t)(k0 + 2 * half) * DD;
            #pragma unroll
            for (int t = 0; t < 8; ++t) {
                v2f bf; bf[0] = w0[t * 16 + lrow]; bf[1] = w0[DD + t * 16 + lrow];
                Hacc[t] = __builtin_amdgcn_wmma_f32_16x16x4_f32(
                    false, a, false, bf, (short)0, Hacc[t], false, false);
            }
        }
    }

    // ---- store H tile
    #pragma unroll
    for (int t = 0; t < 8; ++t)
        #pragma unroll
        for (int v = 0; v < 8; ++v)
            H[((size_t)(nb * 16 + v + 8 * half) * M_NODES + m) * DD + t * 16 + lrow]
                = Hacc[t][v];
}

// ---------------------------------------------------------------------------
// out[r,:] = H[r,:] * dinv[r]^2 (self loop)  + (b0+b1+b2)[:]
// ---------------------------------------------------------------------------
__global__ void k_out_init(const float* __restrict__ H, const float* __restrict__ dinv,
                           const float* __restrict__ b, float* __restrict__ out) {
    int i = blockIdx.x * blockDim.x + threadIdx.x;   // over NM*DD
    int r = i >> 7;
    int d = i & 127;
    float di = dinv[r];
    out[i] = H[i] * di * di + (b[d] + b[DD + d] + b[2 * DD + d]);
}

// ---------------------------------------------------------------------------
// Edge scatter: one wave per edge; lane handles 4 contiguous floats.
// out[dst,:] += H[src,:] * dinv[src]*dinv[dst]
// ---------------------------------------------------------------------------
__global__ __launch_bounds__(256) void k_scatter(
        const float* __restrict__ H, const float* __restrict__ dinv,
        const int* __restrict__ src, const int* __restrict__ dst,
        float* __restrict__ out) {
    int gid  = blockIdx.x * blockDim.x + threadIdx.x;
    int e    = gid >> 5;
    int lane = gid & 31;
    if (e >= NEDGE) return;
    int s = src[e];
    int t = dst[e];
    float norm = dinv[s] * dinv[t];
    float4 hv = *(const float4*)(H + (size_t)s * DD + lane * 4);
    float* op = out + (size_t)t * DD + lane * 4;
    atomicAdd(op + 0, hv.x * norm);
    atomicAdd(op + 1, hv.y * norm);
    atomicAdd(op + 2, hv.z * norm);
    atomicAdd(op + 3, hv.w * norm);
}

// ---------------------------------------------------------------------------
extern "C" void kernel_launch(void* const* d_in, const int* in_sizes, int n_in,
                              void* d_out, int out_size, void* d_ws, size_t ws_size,
                              hipStream_t stream) {
    const float* x    = (const float*)d_in[0];   // [512,256,128]
    const int*   adj  = (const int*)d_in[1];     // [2, NEDGE]
    const float* cadj = (const float*)d_in[2];   // [2,512,512]
    const float* W    = (const float*)d_in[3];   // [3,128,128]
    const float* b    = (const float*)d_in[4];   // [3,128]
    float* out = (float*)d_out;

    float* H   = (float*)d_ws;                   // NM*DD floats (64 MiB)
    float* deg = H + (size_t)NM * DD;            // NM floats (doubles as dinv)

    const int* src = adj;
    const int* dst = adj + NEDGE;

    k_deg_init <<<NM / 256, 256, 0, stream>>>(deg);
    k_deg_edges<<<NEDGE / 256, 256, 0, stream>>>(dst, deg);
    k_dinv     <<<NM / 256, 256, 0, stream>>>(deg);

    k_gemm_fused<<<dim3(M_NODES, N_SET / 16), 32, 0, stream>>>(x, cadj, W, H);

    k_out_init<<<((size_t)NM * DD) / 256, 256, 0, stream>>>(H, deg, b, out);
    k_scatter <<<(NEDGE * 32) / 256, 256, 0, stream>>>(H, deg, src, dst, out);
}